// LFP_33105607917673
// MI455X (gfx1250) — compile-verified
//
#include <hip/hip_runtime.h>

typedef float v2f __attribute__((ext_vector_type(2)));
typedef float v8f __attribute__((ext_vector_type(8)));

#define D_DIM  128
#define K_NB   16
#define BN_EPS 1e-5f
#define MTILES 8     // row tiles of 16 per GEMM block (B regs reused 8x)

// ---------------------------------------------------------------------------
// Kernel 0: zero the BN accumulators (ws is poisoned by harness, never
// re-poisoned between replays, so we must re-init every call).
// stats[0..127] = sum per channel, stats[128..255] = sum of squares.
// ---------------------------------------------------------------------------
__global__ void zero_stats(float* __restrict__ stats) {
    stats[threadIdx.x] = 0.0f;
}

// ---------------------------------------------------------------------------
// Kernel 1: H[m,d] = sum_c X[m,c] * W[d,c]   (M x 128) = (M x 128)(128 x 128)^T
// fp32 WMMA: V_WMMA_F32_16X16X4_F32, 32 k-steps of K=4.
// Grid: M/(16*MTILES) blocks of 256 threads (8 waves). Wave w owns output
// cols [16w, 16w+16) and keeps its 32 B fragments (v2f each, 64 VGPRs)
// resident across MTILES row tiles; X tiles staged in LDS (coalesced float4).
//
// Operand layouts (CDNA5 ISA 7.12.2, 32-bit):
//   A 16x4 : lane L -> row M = L&15 ; VGPR{0,1} -> K = 2*(L>>4) + {0,1}
//   B 4x16 : lane L -> col N = L&15 ; VGPR{0,1} -> K = 2*(L>>4) + {0,1}
//            B(k, d) = W[d, k]  => lane reads W[d0+(L&15)][k0 + 2*(L>>4)+{0,1}]
//   C/D    : lane L, VGPR r -> row r + 8*(L>>4), col L&15
// ---------------------------------------------------------------------------
__global__ void proj_gemm(const float* __restrict__ x,
                          const float* __restrict__ w,
                          float* __restrict__ h) {
    __shared__ float lx[16 * D_DIM];  // 8 KB X tile

    const int tid  = threadIdx.x;
    const int wave = tid >> 5;
    const int lane = tid & 31;
    const int d0   = wave * 16;
    const int mr   = lane & 15;
    const int kh   = (lane >> 4) << 1;   // 0 or 2

    // Load this wave's B fragments once; reused for all MTILES row tiles.
    v2f breg[32];
    const float* wr = w + (size_t)(d0 + mr) * D_DIM + kh;
    #pragma unroll
    for (int kk = 0; kk < 32; ++kk) {
        breg[kk][0] = wr[4 * kk];
        breg[kk][1] = wr[4 * kk + 1];
    }

    for (int t = 0; t < MTILES; ++t) {
        const int m0 = (blockIdx.x * MTILES + t) * 16;

        __syncthreads();   // protect LDS tile reuse across iterations
        const float4* src  = (const float4*)(x + (size_t)m0 * D_DIM);
        float4*       dst4 = (float4*)lx;
        #pragma unroll
        for (int i = tid; i < 16 * D_DIM / 4; i += 256) dst4[i] = src[i];
        __syncthreads();

        v8f acc = {};
        const float* lxr = lx + mr * D_DIM + kh;
        #pragma unroll
        for (int kk = 0; kk < 32; ++kk) {
            v2f a;
            a[0] = lxr[4 * kk];
            a[1] = lxr[4 * kk + 1];
            acc = __builtin_amdgcn_wmma_f32_16x16x4_f32(
                /*neg_a=*/false, a, /*neg_b=*/false, breg[kk],
                /*c_mod=*/(short)0, acc, /*reuse_a=*/false, /*reuse_b=*/false);
        }

        const int mrow = m0 + ((lane >> 4) << 3);
        const int dcol = d0 + mr;
        #pragma unroll
        for (int r = 0; r < 8; ++r)
            h[(size_t)(mrow + r) * D_DIM + dcol] = acc[r];
    }
}

// ---------------------------------------------------------------------------
// Kernel 2: fused spatial-encode + KNN gather + max over K + BN stats.
// One wave per point (strided): lane = group g in [0,32). Each lane owns the
// 4 channels [4g, 4g+4). The per-lane float4 load of h[row][4g..4g+3] makes
// the 512-byte neighbor row fully coalesced across the wave. Gathers are
// L2-resident (h is 16 MB << 192 MB L2). Per-wave partial sums flushed with
// 8 float atomics per lane at the end.
// ---------------------------------------------------------------------------
__global__ void encode_gather_max(const float* __restrict__ h,
                                  const float* __restrict__ xyz,
                                  const int*   __restrict__ knn,
                                  const float* __restrict__ coor,
                                  const float* __restrict__ scale,
                                  float* __restrict__ agg,     // = d_out
                                  float* __restrict__ stats,   // [256] in ws
                                  int Nn, int M) {
    const int lane = threadIdx.x & 31;                       // group index g
    const int wid  = blockIdx.x * (blockDim.x >> 5) + (threadIdx.x >> 5);
    const int nw   = gridDim.x * (blockDim.x >> 5);

    // per-group encoding weights: [cx, cy, cz, scale^2]
    const float cx = coor[3 * lane + 0];
    const float cy = coor[3 * lane + 1];
    const float cz = coor[3 * lane + 2];
    const float sv = scale[lane];
    const float s2 = sv * sv;

    float sm0 = 0.f, sm1 = 0.f, sm2 = 0.f, sm3 = 0.f;   // sum
    float sq0 = 0.f, sq1 = 0.f, sq2 = 0.f, sq3 = 0.f;   // sum of squares

    for (int p = wid; p < M; p += nw) {
        const int b    = p / Nn;
        const int base = b * Nn;
        const float px = xyz[3 * p + 0];
        const float py = xyz[3 * p + 1];
        const float pz = xyz[3 * p + 2];

        float m0 = -3.4e38f, m1 = -3.4e38f, m2 = -3.4e38f, m3 = -3.4e38f;
        const int* kn = knn + (size_t)p * K_NB;
        #pragma unroll 4
        for (int k = 0; k < K_NB; ++k) {
            const int j = base + kn[k];
            const float rx = xyz[3 * j + 0] - px;
            const float ry = xyz[3 * j + 1] - py;
            const float rz = xyz[3 * j + 2] - pz;
            const float e  = cx * rx + cy * ry + cz * rz
                           + s2 * (rx * rx + ry * ry + rz * rz);
            const float4 hv = *(const float4*)(h + (size_t)j * D_DIM + 4 * lane);
            m0 = fmaxf(m0, hv.x + e);
            m1 = fmaxf(m1, hv.y + e);
            m2 = fmaxf(m2, hv.z + e);
            m3 = fmaxf(m3, hv.w + e);
        }
        float4 outv = make_float4(m0, m1, m2, m3);
        *(float4*)(agg + (size_t)p * D_DIM + 4 * lane) = outv;

        sm0 += m0; sm1 += m1; sm2 += m2; sm3 += m3;
        sq0 += m0 * m0; sq1 += m1 * m1; sq2 += m2 * m2; sq3 += m3 * m3;
    }

    atomicAdd(&stats[4 * lane + 0], sm0);
    atomicAdd(&stats[4 * lane + 1], sm1);
    atomicAdd(&stats[4 * lane + 2], sm2);
    atomicAdd(&stats[4 * lane + 3], sm3);
    atomicAdd(&stats[128 + 4 * lane + 0], sq0);
    atomicAdd(&stats[128 + 4 * lane + 1], sq1);
    atomicAdd(&stats[128 + 4 * lane + 2], sq2);
    atomicAdd(&stats[128 + 4 * lane + 3], sq3);
}

// ---------------------------------------------------------------------------
// Kernel 3: BatchNorm apply, in place on d_out.
// mean = S/M ; var = Q/M - mean^2 (biased, matches reference).
// ---------------------------------------------------------------------------
__global__ void bn_apply(float* __restrict__ out,
                         const float* __restrict__ stats,
                         const float* __restrict__ bw,
                         const float* __restrict__ bb,
                         float invM, int total) {
    const int idx = blockIdx.x * blockDim.x + threadIdx.x;
    if (idx >= total) return;
    const int c = idx & (D_DIM - 1);
    const float mean = stats[c] * invM;
    const float var  = stats[D_DIM + c] * invM - mean * mean;
    out[idx] = (out[idx] - mean) * rsqrtf(var + BN_EPS) * bw[c] + bb[c];
}

// ---------------------------------------------------------------------------
extern "C" void kernel_launch(void* const* d_in, const int* in_sizes, int n_in,
                              void* d_out, int out_size, void* d_ws, size_t ws_size,
                              hipStream_t stream) {
    const float* x      = (const float*)d_in[0];  // [B,N,128]
    const float* xyz    = (const float*)d_in[1];  // [B,N,3]
    const int*   knn    = (const int*)  d_in[2];  // [B,N,16]
    const float* proj_w = (const float*)d_in[3];  // [128,128]
    const float* coor   = (const float*)d_in[4];  // [96]
    const float* scale  = (const float*)d_in[5];  // [32]
    const float* bnw    = (const float*)d_in[6];  // [128]
    const float* bnb    = (const float*)d_in[7];  // [128]
    float*       out    = (float*)d_out;

    const int M = in_sizes[0] / D_DIM;  // B*N = 32768
    const int B = 4;
    const int N = M / B;

    float* stats = (float*)d_ws;          // 256 floats
    float* h     = (float*)d_ws + 256;    // M*128 floats (16 MB)

    zero_stats<<<1, 256, 0, stream>>>(stats);

    proj_gemm<<<M / (16 * MTILES), 256, 0, stream>>>(x, proj_w, h);

    encode_gather_max<<<256, 256, 0, stream>>>(h, xyz, knn, coor, scale,
                                               out, stats, N, M);

    const int total = M * D_DIM;
    bn_apply<<<(total + 255) / 256, 256, 0, stream>>>(
        out, stats, bnw, bnb, 1.0f / (float)M, total);
}